// GCNConv_5342939316956
// MI455X (gfx1250) — compile-verified
//
#include <hip/hip_runtime.h>
#include <hip/hip_bf16.h>

typedef __attribute__((ext_vector_type(2))) float v2f;
typedef __attribute__((ext_vector_type(8))) float v8f;

#define N_NODES 100000
#define N_EDGES 1600000
#define D 128           // D_IN == D_OUT == 128

// ---------------------------------------------------------------------------
// Kernel 0: zero the output (harness poisons d_out; segment-sum needs zeros)
// ---------------------------------------------------------------------------
__global__ __launch_bounds__(256) void zero_out(float* __restrict__ out, int n4) {
    int i = blockIdx.x * blockDim.x + threadIdx.x;
    if (i < n4) {
        ((float4*)out)[i] = make_float4(0.f, 0.f, 0.f, 0.f);
    }
}

// ---------------------------------------------------------------------------
// Kernel 1: h = x @ W^T  via V_WMMA_F32_16X16X4_F32
//   x: [N_NODES, D] row-major, W: [D, D] row-major (h[n][o] = sum_k x[n][k]*W[o][k])
//   grid.x = N_NODES/16 row tiles; block = (32, 8): wave w owns column tile w.
//
// A (16x4 f32) per-lane layout: lane L holds row L%16;
//   VGPR0 = K = k0 + (L<16 ? 0 : 2), VGPR1 = next K  -> one float2 load.
// B (4x16 f32): B[k][n] = W[n][k]; mirrored layout -> one float2 load from W row.
// C/D (16x16 f32, 8 VGPRs): VGPR v holds row v (lanes 0-15) / row v+8 (lanes 16-31),
//   column = lane % 16.
// ---------------------------------------------------------------------------
__global__ __launch_bounds__(256) void gemm_xWt(const float* __restrict__ x,
                                                const float* __restrict__ W,
                                                float* __restrict__ h) {
    const int lane    = threadIdx.x;            // 0..31
    const int wave    = threadIdx.y;            // 0..7 -> column tile
    const int rowBase = blockIdx.x * 16;        // N_NODES = 6250 * 16 exactly
    const int colBase = wave * 16;

    const int l16  = lane & 15;
    const int koff = (lane >> 4) << 1;          // 0 for lanes 0-15, 2 for 16-31

    const float* arow = x + (size_t)(rowBase + l16) * D;  // A row (x)
    const float* brow = W + (size_t)(colBase + l16) * D;  // B^T row (W[col][*])

    v8f c = {0.f, 0.f, 0.f, 0.f, 0.f, 0.f, 0.f, 0.f};

#pragma unroll
    for (int k0 = 0; k0 < D; k0 += 4) {
        v2f a = *(const v2f*)(arow + k0 + koff);
        v2f b = *(const v2f*)(brow + k0 + koff);
        // (neg_a, A, neg_b, B, c_mod, C, reuse_a, reuse_b)
        c = __builtin_amdgcn_wmma_f32_16x16x4_f32(false, a, false, b,
                                                  (short)0, c, false, false);
    }

    const int rhalf = (lane >> 4) * 8;          // lanes 16-31 hold rows +8
#pragma unroll
    for (int v = 0; v < 8; ++v) {
        h[(size_t)(rowBase + v + rhalf) * D + colBase + l16] = c[v];
    }
}

// ---------------------------------------------------------------------------
// Kernel 2: out[rows[e]] += vals[e] * h[cols[e]]   (segment-sum via L2 atomics)
//   One wave per edge: lane holds float4 -> 32 lanes * 16B = full 512B row,
//   perfectly coalesced gather, then 4 global_atomic_add_f32 per lane.
// ---------------------------------------------------------------------------
__global__ __launch_bounds__(256) void aggregate(const float* __restrict__ h,
                                                 const int* __restrict__ rows,
                                                 const int* __restrict__ cols,
                                                 const float* __restrict__ vals,
                                                 float* __restrict__ out) {
    const int lane = threadIdx.x & 31;
    const int edge = (blockIdx.x * blockDim.x + threadIdx.x) >> 5;
    if (edge >= N_EDGES) return;

    const int   r = rows[edge];
    const int   s = cols[edge];
    const float v = vals[edge];

    const float4 hv = *(const float4*)(h + (size_t)s * D + lane * 4);
    float* o = out + (size_t)r * D + lane * 4;

    atomicAdd(o + 0, v * hv.x);
    atomicAdd(o + 1, v * hv.y);
    atomicAdd(o + 2, v * hv.z);
    atomicAdd(o + 3, v * hv.w);
}

// ---------------------------------------------------------------------------
// Launch
//   d_in: [x (N*D f32), W (D*D f32), adj_rows (E i32), adj_cols (E i32),
//          adj_vals (E f32)]
//   d_out: [N*D f32], d_ws: scratch for h (N*D*4 = 51.2 MB)
// ---------------------------------------------------------------------------
extern "C" void kernel_launch(void* const* d_in, const int* in_sizes, int n_in,
                              void* d_out, int out_size, void* d_ws, size_t ws_size,
                              hipStream_t stream) {
    const float* x    = (const float*)d_in[0];
    const float* W    = (const float*)d_in[1];
    const int*   rows = (const int*)d_in[2];
    const int*   cols = (const int*)d_in[3];
    const float* vals = (const float*)d_in[4];
    float*       out  = (float*)d_out;
    float*       h    = (float*)d_ws;   // [N_NODES, D] projected features

    // 0) zero output
    {
        int n4 = (N_NODES * D) / 4;                 // 3,200,000 float4s
        int blocks = (n4 + 255) / 256;
        zero_out<<<blocks, 256, 0, stream>>>(out, n4);
    }

    // 1) dense projection h = x @ W^T  (WMMA f32 16x16x4)
    {
        dim3 block(32, 8);                          // 8 waves = 8 column tiles
        dim3 grid(N_NODES / 16);                    // 6250 row tiles
        gemm_xWt<<<grid, block, 0, stream>>>(x, W, h);
    }

    // 2) sparse aggregation (one wave per edge)
    {
        long long threads = (long long)N_EDGES * 32;
        int blocks = (int)((threads + 255) / 256);  // 200,000 blocks
        aggregate<<<blocks, 256, 0, stream>>>(h, rows, cols, vals, out);
    }
}